// pointconvbackbonetest_75325136437736
// MI455X (gfx1250) — compile-verified
//
#include <hip/hip_runtime.h>

#define NPTS 8192
#define NSAMP 32
#define BWID 0.1f

typedef __attribute__((ext_vector_type(2))) float v2f;
typedef __attribute__((ext_vector_type(8))) float v8f;

// ---------------------------------------------------------------------------
// Wave-level 16x16 f32 tile GEMM using V_WMMA_F32_16X16X4_F32.
// A: MxK row-major (lda), B: KxN row-major (ldb), both in LDS. K % 4 == 0.
// Fragment layouts per CDNA5 ISA 7.12.2:
//   A 16x4 : lanes 0-15 hold M=lane, VGPR0=K0,VGPR1=K1; lanes 16-31 K2/K3
//   B 4x16 : VGPR v holds row K=v (lanes 0-15) / K=v+2 (lanes 16-31), N=lane&15
//   C/D    : VGPR v holds row M=v (lanes 0-15) / M=v+8 (lanes 16-31), N=lane&15
// ---------------------------------------------------------------------------
__device__ __forceinline__ v8f wmma_f32_tile(const float* __restrict__ A, int lda,
                                             const float* __restrict__ B, int ldb,
                                             int K, v8f acc) {
  const int lane = threadIdx.x & 31;
  const int mn = lane & 15;
  const int kh = (lane >> 4) << 1; // 0 for lanes 0-15, 2 for lanes 16-31
  for (int kb = 0; kb < K; kb += 4) {
    v2f a, b;
    a.x = A[mn * lda + kb + kh];
    a.y = A[mn * lda + kb + kh + 1];
    b.x = B[(kb + kh) * ldb + mn];
    b.y = B[(kb + kh + 1) * ldb + mn];
    acc = __builtin_amdgcn_wmma_f32_16x16x4_f32(false, a, false, b,
                                                (short)0, acc, false, false);
  }
  return acc;
}

// ---------------------------------------------------------------------------
// Kernel 1: per batch, d^2 to point 0, stable 32-NN selection (matches
// jax.lax.top_k(-d2, 32): ascending d2, ties broken by lower index).
// ---------------------------------------------------------------------------
__global__ void __launch_bounds__(256) topk_kernel(const float* __restrict__ xyz,
                                                   int* __restrict__ idx_sel) {
  const int b = blockIdx.x;
  const int tid = threadIdx.x;
  __shared__ float sD2[NPTS];
  __shared__ float rv[256];
  __shared__ int ri[256];
  const float* xb = xyz + (size_t)b * 3 * NPTS;
  const float x0 = xb[0], y0 = xb[NPTS], z0 = xb[2 * NPTS];
  for (int n = tid; n < NPTS; n += 256) {
    float dx = xb[n] - x0, dy = xb[NPTS + n] - y0, dz = xb[2 * NPTS + n] - z0;
    sD2[n] = dx * dx + dy * dy + dz * dz;
  }
  __syncthreads();
  for (int s = 0; s < NSAMP; ++s) {
    float bv = 3.4e38f;
    int bi = NPTS;
    for (int n = tid; n < NPTS; n += 256) {
      float v = sD2[n];
      if (v < bv) { bv = v; bi = n; }   // strictly-less: lower n wins ties in-thread
    }
    rv[tid] = bv; ri[tid] = bi;
    __syncthreads();
    for (int off = 128; off > 0; off >>= 1) {
      if (tid < off) {
        float ov = rv[tid + off]; int oi = ri[tid + off];
        if (ov < rv[tid] || (ov == rv[tid] && oi < ri[tid])) { rv[tid] = ov; ri[tid] = oi; }
      }
      __syncthreads();
    }
    if (tid == 0) {
      idx_sel[b * NSAMP + s] = ri[0];
      sD2[ri[0]] = 3.4e38f;  // exclude from further rounds
    }
    __syncthreads();
  }
}

// ---------------------------------------------------------------------------
// Kernel 2: density only at the 64 selected points (the reference's 8192x8192
// density matrix is only ever read at the gathered indices).
// inv_d = (2.5*BW*NPTS) / sum_m exp(-||xi-xm||^2/(2*BW^2))
// ---------------------------------------------------------------------------
__global__ void __launch_bounds__(256) density_kernel(const float* __restrict__ xyz,
                                                      const int* __restrict__ idx_sel,
                                                      float* __restrict__ inv_d) {
  const int s = blockIdx.x;  // 0..31
  const int b = blockIdx.y;  // 0..1
  const int tid = threadIdx.x;
  __shared__ float red[256];
  const float* xb = xyz + (size_t)b * 3 * NPTS;
  const int i = idx_sel[b * NSAMP + s];
  const float xi = xb[i], yi = xb[NPTS + i], zi = xb[2 * NPTS + i];
  const float c = -1.0f / (2.0f * BWID * BWID);  // -50
  float sum = 0.f;
  for (int m = tid; m < NPTS; m += 256) {
    if (m + 256 < NPTS) __builtin_prefetch(xb + m + 256, 0, 0);  // global_prefetch_b8
    float dx = xb[m] - xi, dy = xb[NPTS + m] - yi, dz = xb[2 * NPTS + m] - zi;
    float d2 = dx * dx + dy * dy + dz * dz;
    sum += __expf(d2 * c);
  }
  red[tid] = sum;
  __syncthreads();
  for (int off = 128; off > 0; off >>= 1) {
    if (tid < off) red[tid] += red[tid + off];
    __syncthreads();
  }
  if (tid == 0) inv_d[b * NSAMP + s] = (2.5f * BWID * (float)NPTS) / red[0];
}

// ---------------------------------------------------------------------------
// Kernel 3: gathers + all MLPs + aggregation + final projection.
// One block of 128 threads (4 wave32) per batch; GEMMs via f32 WMMA.
// ---------------------------------------------------------------------------
__global__ void __launch_bounds__(128) mlp_kernel(
    const float* __restrict__ xyz, const float* __restrict__ feat,
    const float* __restrict__ w1, const float* __restrict__ b1,
    const float* __restrict__ g1, const float* __restrict__ be1,
    const float* __restrict__ w2, const float* __restrict__ b2,
    const float* __restrict__ g2, const float* __restrict__ be2,
    const float* __restrict__ w3, const float* __restrict__ b3,
    const float* __restrict__ g3, const float* __restrict__ be3,
    const float* __restrict__ ww1, const float* __restrict__ wb1,
    const float* __restrict__ wg1, const float* __restrict__ wbe1,
    const float* __restrict__ ww2, const float* __restrict__ wb2,
    const float* __restrict__ wg2, const float* __restrict__ wbe2,
    const float* __restrict__ ww3, const float* __restrict__ wb3,
    const float* __restrict__ wg3, const float* __restrict__ wbe3,
    const float* __restrict__ dw1, const float* __restrict__ db1,
    const float* __restrict__ dg1, const float* __restrict__ dbe1,
    const float* __restrict__ dw2, const float* __restrict__ db2,
    const float* __restrict__ dg2, const float* __restrict__ dbe2,
    const float* __restrict__ dw3, const float* __restrict__ db3,
    const float* __restrict__ dg3, const float* __restrict__ dbe3,
    const float* __restrict__ lw, const float* __restrict__ lb,
    const float* __restrict__ lg, const float* __restrict__ lbe,
    const int* __restrict__ idx_sel, const float* __restrict__ inv_d,
    float* __restrict__ out) {
  const int b = blockIdx.x;
  const int tid = threadIdx.x;
  const int wave = tid >> 5;
  const int lane = tid & 31;

  // Static LDS pool (~59 KB)
  __shared__ float smem[15168];
  float* sB   = smem;            // 4096: staged transposed weights (64x64 max)
  float* sXa  = smem + 4096;     // 2048: 32x64 activations (L1 out)
  float* sXb  = smem + 6144;     // 2048: 32x64 activations (L2 out)
  float* sPT  = smem + 8192;     // 4096: p transposed, 128x32
  float* sPts = smem + 12288;    // 256 : 32x8 (concat[g_norm,g_feat], K padded 6->8)
  float* sWm  = smem + 12544;    // 512 : 32x16 weight-net output
  float* sDs  = smem + 13056;    // 32  : density scale per point
  float* sAgg = smem + 13088;    // 2048: 128x16 aggregation
  float* sRed = smem + 15136;    // 32  : scratch

  const float* xb = xyz + (size_t)b * 3 * NPTS;
  const float* fb = feat + (size_t)b * 3 * NPTS;

  // ---- gathers ----
  if (tid < 32) {
    const int i = idx_sel[b * NSAMP + tid];
    sPts[tid * 8 + 0] = xb[i] - xb[0];
    sPts[tid * 8 + 1] = xb[NPTS + i] - xb[NPTS];
    sPts[tid * 8 + 2] = xb[2 * NPTS + i] - xb[2 * NPTS];
    sPts[tid * 8 + 3] = fb[i];
    sPts[tid * 8 + 4] = fb[NPTS + i];
    sPts[tid * 8 + 5] = fb[2 * NPTS + i];
    sPts[tid * 8 + 6] = 0.f;
    sPts[tid * 8 + 7] = 0.f;
    sRed[tid] = inv_d[b * NSAMP + tid];
  }
  __syncthreads();

  // ---- density-scale MLP (1->16->8->1, sigmoid) and weight-net (3->8->8->16), scalar ----
  if (tid < 32) {
    float mx = sRed[0];
    for (int j = 1; j < 32; ++j) mx = fmaxf(mx, sRed[j]);
    const float dsn = sRed[tid] / mx;
    float h1[16];
    for (int o = 0; o < 16; ++o) {
      float y = dsn * dw1[o] + db1[o];
      h1[o] = fmaxf(dg1[o] * y + dbe1[o], 0.f);
    }
    float h2[8];
    for (int o = 0; o < 8; ++o) {
      float y = db2[o];
      for (int k = 0; k < 16; ++k) y += h1[k] * dw2[o * 16 + k];
      h2[o] = fmaxf(dg2[o] * y + dbe2[o], 0.f);
    }
    float y3 = db3[0];
    for (int k = 0; k < 8; ++k) y3 += h2[k] * dw3[k];
    y3 = dg3[0] * y3 + dbe3[0];
    sDs[tid] = 1.f / (1.f + __expf(-y3));

    float a1[8];
    for (int o = 0; o < 8; ++o) {
      float t = wb1[o] + sPts[tid * 8 + 0] * ww1[o * 3 + 0] +
                sPts[tid * 8 + 1] * ww1[o * 3 + 1] +
                sPts[tid * 8 + 2] * ww1[o * 3 + 2];
      a1[o] = fmaxf(wg1[o] * t + wbe1[o], 0.f);
    }
    float a2[8];
    for (int o = 0; o < 8; ++o) {
      float t = wb2[o];
      for (int k = 0; k < 8; ++k) t += a1[k] * ww2[o * 8 + k];
      a2[o] = fmaxf(wg2[o] * t + wbe2[o], 0.f);
    }
    for (int o = 0; o < 16; ++o) {
      float t = wb3[o];
      for (int k = 0; k < 8; ++k) t += a2[k] * ww3[o * 8 + k];
      sWm[tid * 16 + o] = fmaxf(wg3[o] * t + wbe3[o], 0.f);
    }
  }

  // ---- L1: (32x8) @ (8x64), K padded 6->8 ----
  for (int p = tid; p < 8 * 64; p += 128) {
    int k = p >> 6, o = p & 63;
    sB[p] = (k < 6) ? w1[o * 6 + k] : 0.f;
  }
  __syncthreads();
  for (int t = wave; t < 8; t += 4) {
    int mi = t >> 2, ni = t & 3;
    v8f acc = {};
    acc = wmma_f32_tile(sPts + mi * 16 * 8, 8, sB + ni * 16, 64, 8, acc);
#pragma unroll
    for (int v = 0; v < 8; ++v) {
      int row = mi * 16 + v + ((lane >> 4) << 3);
      int col = ni * 16 + (lane & 15);
      float yv = acc[v] + b1[col];
      sXa[row * 64 + col] = fmaxf(g1[col] * yv + be1[col], 0.f);
    }
  }
  __syncthreads();

  // ---- L2: (32x64) @ (64x64) ----
  for (int p = tid; p < 64 * 64; p += 128) {
    int k = p >> 6, o = p & 63;
    sB[p] = w2[o * 64 + k];
  }
  __syncthreads();
  for (int t = wave; t < 8; t += 4) {
    int mi = t >> 2, ni = t & 3;
    v8f acc = {};
    acc = wmma_f32_tile(sXa + mi * 16 * 64, 64, sB + ni * 16, 64, 64, acc);
#pragma unroll
    for (int v = 0; v < 8; ++v) {
      int row = mi * 16 + v + ((lane >> 4) << 3);
      int col = ni * 16 + (lane & 15);
      float yv = acc[v] + b2[col];
      sXb[row * 64 + col] = fmaxf(g2[col] * yv + be2[col], 0.f);
    }
  }

  // ---- L3: (32x64) @ (64x128) in two 64-col halves; epilogue: CBR * ds, store transposed ----
  for (int h = 0; h < 2; ++h) {
    __syncthreads();  // previous consumers of sB done
    for (int p = tid; p < 64 * 64; p += 128) {
      int k = p >> 6, o = p & 63;
      sB[p] = w3[(h * 64 + o) * 64 + k];
    }
    __syncthreads();
    for (int t = wave; t < 8; t += 4) {
      int mi = t >> 2, ni = t & 3;
      v8f acc = {};
      acc = wmma_f32_tile(sXb + mi * 16 * 64, 64, sB + ni * 16, 64, 64, acc);
#pragma unroll
      for (int v = 0; v < 8; ++v) {
        int row = mi * 16 + v + ((lane >> 4) << 3);
        int col = h * 64 + ni * 16 + (lane & 15);
        float yv = acc[v] + b3[col];
        float pv = fmaxf(g3[col] * yv + be3[col], 0.f) * sDs[row];
        sPT[col * 32 + row] = pv;  // p^T for aggregation
      }
    }
  }
  __syncthreads();

  // ---- aggregation: (128x32) @ (32x16) = p^T w ----
  for (int mi = wave; mi < 8; mi += 4) {
    v8f acc = {};
    acc = wmma_f32_tile(sPT + mi * 16 * 32, 32, sWm, 16, 32, acc);
#pragma unroll
    for (int v = 0; v < 8; ++v) {
      int row = mi * 16 + v + ((lane >> 4) << 3);
      int col = lane & 15;
      sAgg[row * 16 + col] = acc[v];  // flat index c*16+k matches reshape order
    }
  }
  __syncthreads();

  // ---- final: out[o] = relu(lg*(agg . lw[o] + lb) + lbe), o = 0..127 ----
  {
    const int o = tid;
    const float* lwo = lw + (size_t)o * 2048;
    float accv = 0.f;
    for (int j = 0; j < 2048; ++j) accv += sAgg[j] * lwo[j];
    accv += lb[o];
    out[b * 128 + o] = fmaxf(lg[o] * accv + lbe[o], 0.f);
  }
}

extern "C" void kernel_launch(void* const* d_in, const int* in_sizes, int n_in,
                              void* d_out, int out_size, void* d_ws, size_t ws_size,
                              hipStream_t stream) {
  const float* xyz = (const float*)d_in[0];
  const float* feat = (const float*)d_in[1];
  const float* w1 = (const float*)d_in[2];
  const float* b1 = (const float*)d_in[3];
  const float* g1 = (const float*)d_in[4];
  const float* be1 = (const float*)d_in[5];
  const float* w2 = (const float*)d_in[6];
  const float* b2 = (const float*)d_in[7];
  const float* g2 = (const float*)d_in[8];
  const float* be2 = (const float*)d_in[9];
  const float* w3 = (const float*)d_in[10];
  const float* b3 = (const float*)d_in[11];
  const float* g3 = (const float*)d_in[12];
  const float* be3 = (const float*)d_in[13];
  const float* ww1 = (const float*)d_in[14];
  const float* wb1 = (const float*)d_in[15];
  const float* wg1 = (const float*)d_in[16];
  const float* wbe1 = (const float*)d_in[17];
  const float* ww2 = (const float*)d_in[18];
  const float* wb2 = (const float*)d_in[19];
  const float* wg2 = (const float*)d_in[20];
  const float* wbe2 = (const float*)d_in[21];
  const float* ww3 = (const float*)d_in[22];
  const float* wb3 = (const float*)d_in[23];
  const float* wg3 = (const float*)d_in[24];
  const float* wbe3 = (const float*)d_in[25];
  const float* dw1 = (const float*)d_in[26];
  const float* db1 = (const float*)d_in[27];
  const float* dg1 = (const float*)d_in[28];
  const float* dbe1 = (const float*)d_in[29];
  const float* dw2 = (const float*)d_in[30];
  const float* db2 = (const float*)d_in[31];
  const float* dg2 = (const float*)d_in[32];
  const float* dbe2 = (const float*)d_in[33];
  const float* dw3 = (const float*)d_in[34];
  const float* db3 = (const float*)d_in[35];
  const float* dg3 = (const float*)d_in[36];
  const float* dbe3 = (const float*)d_in[37];
  const float* lw = (const float*)d_in[38];
  const float* lb = (const float*)d_in[39];
  const float* lg = (const float*)d_in[40];
  const float* lbe = (const float*)d_in[41];

  int* idx_sel = (int*)d_ws;                          // 64 ints
  float* inv_d = (float*)((char*)d_ws + 256);         // 64 floats
  float* out = (float*)d_out;                         // (2,128)

  topk_kernel<<<2, 256, 0, stream>>>(xyz, idx_sel);
  density_kernel<<<dim3(NSAMP, 2), 256, 0, stream>>>(xyz, idx_sel, inv_d);
  mlp_kernel<<<2, 128, 0, stream>>>(
      xyz, feat, w1, b1, g1, be1, w2, b2, g2, be2, w3, b3, g3, be3,
      ww1, wb1, wg1, wbe1, ww2, wb2, wg2, wbe2, ww3, wb3, wg3, wbe3,
      dw1, db1, dg1, dbe1, dw2, db2, dg2, dbe2, dw3, db3, dg3, dbe3,
      lw, lb, lg, lbe, idx_sel, inv_d, out);
}